// GenLayer_57191784514055
// MI455X (gfx1250) — compile-verified
//
#include <hip/hip_runtime.h>
#include <stdint.h>

typedef __attribute__((ext_vector_type(2))) float v2f;
typedef __attribute__((ext_vector_type(8))) float v8f;
typedef int v4i_vs __attribute__((vector_size(16)));   // matches builtin param type

#define D_MODEL  512
#define PRED_LEN 720
#define CHANNELS 321
#define BATCH    64

#define LDS_STRIDE 516              // 512 + 4 pad -> conflict-free ds_load_b64
#define PBLOCK     128              // 8 waves * 16 cols
#define NPB        6                // ceil(720 / 128)

// ---------------------------------------------------------------------------
// out[b,c,p] = sum_d x[b,c,d] * W[c,d,p] + bias[c,p]
// One workgroup = one (channel, 128-col P block). X_c staged in LDS (async
// global->LDS path), W streamed nontemporally, fp32 WMMA 16x16x4 accumulation.
// ---------------------------------------------------------------------------
extern "C" __global__ __launch_bounds__(256)
void chanlinear_wmma_f32(const float* __restrict__ X,
                         const float* __restrict__ W,
                         const float* __restrict__ Bias,
                         float* __restrict__ Out)
{
    extern __shared__ float smem[];            // 64 * 516 * 4B = 132096 B

    const int c   = blockIdx.x / NPB;
    const int pb  = blockIdx.x % NPB;
    const int tid = threadIdx.x;

    // ---- Stage X[:, c, :] (64 rows x 512 f32, row stride C*D in global) ----
    // 8192 16B-chunks total, 32 per thread.
#if __has_builtin(__builtin_amdgcn_global_load_async_to_lds_b128)
    #pragma unroll 4
    for (int i = 0; i < 32; ++i) {
        const int chunk = tid + (i << 8);
        const int row   = chunk >> 7;
        const int cx    = (chunk & 127) << 2;
        const float* g  = X + ((size_t)row * CHANNELS + c) * D_MODEL + cx;
        float*       l  = &smem[row * LDS_STRIDE + cx];
        __builtin_amdgcn_global_load_async_to_lds_b128(
            (v4i_vs*)(const_cast<float*>(g)),
            (v4i_vs*)l,
            0, 0);
    }
    asm volatile("s_wait_asynccnt 0" ::: "memory");
#else
    #pragma unroll 4
    for (int i = 0; i < 32; ++i) {
        const int chunk = tid + (i << 8);
        const int row   = chunk >> 7;
        const int cx    = (chunk & 127) << 2;
        const float4* g = (const float4*)(X + ((size_t)row * CHANNELS + c) * D_MODEL + cx);
        *(float4*)&smem[row * LDS_STRIDE + cx] = *g;
    }
#endif
    __syncthreads();

    const int wave = tid >> 5;
    const int lane = tid & 31;
    const int n0   = pb * PBLOCK + wave * 16;
    if (n0 >= PRED_LEN) return;                 // wave-uniform: EXEC stays all-1s

    const int col   = n0 + (lane & 15);
    const int rowL  = lane & 15;                // M within tile
    const int khalf = (lane >> 4) << 1;         // upper lane half carries K+2,K+3

    const float* Wc = W + (size_t)c * D_MODEL * PRED_LEN + col;

    v8f acc0 = {}, acc1 = {}, acc2 = {}, acc3 = {};

    const float* a0p = &smem[(rowL +  0) * LDS_STRIDE];
    const float* a1p = &smem[(rowL + 16) * LDS_STRIDE];
    const float* a2p = &smem[(rowL + 32) * LDS_STRIDE];
    const float* a3p = &smem[(rowL + 48) * LDS_STRIDE];

    #pragma unroll 4
    for (int k = 0; k < D_MODEL; k += 4) {
        const int kk = k + khalf;
        // B fragment: 4x16, lane half selects K rows (mirrors A layout)
        v2f bf;
        bf.x = __builtin_nontemporal_load(Wc + (size_t)kk * PRED_LEN);
        bf.y = __builtin_nontemporal_load(Wc + (size_t)(kk + 1) * PRED_LEN);
        // A fragments: one ds_load_b64 each (kk even -> 8B aligned, padded banks)
        v2f a0 = *(const v2f*)(a0p + kk);
        v2f a1 = *(const v2f*)(a1p + kk);
        v2f a2 = *(const v2f*)(a2p + kk);
        v2f a3 = *(const v2f*)(a3p + kk);

        acc0 = __builtin_amdgcn_wmma_f32_16x16x4_f32(false, a0, false, bf, (short)0, acc0, false, false);
        acc1 = __builtin_amdgcn_wmma_f32_16x16x4_f32(false, a1, false, bf, (short)0, acc1, false, false);
        acc2 = __builtin_amdgcn_wmma_f32_16x16x4_f32(false, a2, false, bf, (short)0, acc2, false, false);
        acc3 = __builtin_amdgcn_wmma_f32_16x16x4_f32(false, a3, false, bf, (short)0, acc3, false, false);
    }

    // ---- bias + store: C/D layout: VGPR j, lanes 0-15 -> M=j, lanes 16-31 -> M=j+8
    const float bv   = Bias[(size_t)c * PRED_LEN + col];
    const int   mrow = (lane >> 4) << 3;        // 0 or 8

    v8f acc[4] = { acc0, acc1, acc2, acc3 };
    #pragma unroll
    for (int m = 0; m < 4; ++m) {
        #pragma unroll
        for (int j = 0; j < 8; ++j) {
            const int b = m * 16 + mrow + j;
            __builtin_nontemporal_store(acc[m][j] + bv,
                Out + ((size_t)b * CHANNELS + c) * PRED_LEN + col);
        }
    }
}

extern "C" void kernel_launch(void* const* d_in, const int* in_sizes, int n_in,
                              void* d_out, int out_size, void* d_ws, size_t ws_size,
                              hipStream_t stream) {
    const float* x  = (const float*)d_in[0];
    const float* W  = (const float*)d_in[1];
    const float* b  = (const float*)d_in[2];
    float* out      = (float*)d_out;

    const dim3 grid(CHANNELS * NPB);
    const dim3 block(256);
    const size_t shmem = (size_t)BATCH * LDS_STRIDE * sizeof(float); // 132096 B
    hipLaunchKernelGGL(chanlinear_wmma_f32, grid, block, shmem, stream,
                       x, W, b, out);
}